// PersonAttr_Loss_13993003451064
// MI455X (gfx1250) — compile-verified
//
#include <hip/hip_runtime.h>
#include <stdint.h>

// ---------------------------------------------------------------------------
// PersonAttr multi-head loss, MI455X (gfx1250).
// Memory-bound streaming reduction: 252 MB read -> 2 floats.
// CDNA5 paths used: global_load_async_to_lds_b128 (ASYNCcnt DMA, double
// buffered), s_wait_asynccnt, NT-hinted b128 vector loads, wave32 shfl
// reductions. Deterministic two-pass final reduction (no float atomics).
// ---------------------------------------------------------------------------

#define EPSF 1e-6f

constexpr int BLOCK      = 256;
constexpr int TILE_ROWS  = 256;                 // rows per tile (== BLOCK)
constexpr int TGT_INTS   = TILE_ROWS * 15;      // 3840 ints  = 15360 bytes
constexpr int TGT_CHUNKS = TGT_INTS / 4;        // 960 x 16B chunks (16B aligned)
constexpr int MAX_GRID   = 2048;

typedef float v4f __attribute__((ext_vector_type(4)));

__device__ __forceinline__ float waveRedF(float v) {
#pragma unroll
  for (int o = 16; o > 0; o >>= 1) v += __shfl_xor(v, o, 32);
  return v;
}
__device__ __forceinline__ unsigned waveRedU(unsigned v) {
#pragma unroll
  for (int o = 16; o > 0; o >>= 1) v += __shfl_xor(v, o, 32);
  return v;
}

__device__ __forceinline__ void asyncWait0() {
#if __has_builtin(__builtin_amdgcn_s_wait_asynccnt)
  __builtin_amdgcn_s_wait_asynccnt(0);
#else
  asm volatile("s_wait_asynccnt 0" ::: "memory");
#endif
}

// GVS-mode async DMA: copies 16 bytes per lane from global address base plus goff to LDS offset lds_off.
__device__ __forceinline__ void asyncLoadB128(uint32_t lds_off, uint32_t goff,
                                              const void* base) {
  asm volatile("global_load_async_to_lds_b128 %0, %1, %2"
               :
               : "v"(lds_off), "v"(goff),
                 "s"((unsigned long long)(uintptr_t)base)
               : "memory");
}

__device__ __forceinline__ void issueTile(const int* __restrict__ targets,
                                          int tile, uint32_t ldsBufBase,
                                          int tid) {
  const int* src = targets + (size_t)tile * TGT_INTS;  // 16B-aligned tile base
#pragma unroll
  for (int j = 0; j < 4; ++j) {                        // 960 chunks / 256 thr
    int idx = tid + j * BLOCK;
    if (idx < TGT_CHUNKS)
      asyncLoadB128(ldsBufBase + (uint32_t)idx * 16u, (uint32_t)idx * 16u, src);
  }
}

// _clc_loss_mask contribution + success for one binary head element.
__device__ __forceinline__ void binHead(float x, int g, float pw,
                                        float& s, unsigned& succ) {
  float p = 1.f / (1.f + expf(-x));
  p = fminf(fmaxf(p, EPSF), 1.f - EPSF);
  float q = g ? (1.f - p) : p;          // (1-p) on pos, p on neg
  float l = logf(g ? p : (1.f - p));
  float w = g ? pw : (1.f - pw);
  s += l * q * q * w;
  succ += ((((p > 0.5f) ? 1 : 0) == g) ? 1u : 0u);
}

__global__ __launch_bounds__(BLOCK) void pa_loss_main(
    const float* __restrict__ out0, const float* __restrict__ out1,
    const float* __restrict__ out2, const float* __restrict__ out3,
    const float* __restrict__ out4, const float* __restrict__ pos_dist,
    const int* __restrict__ targets,
    float* __restrict__ wsF, unsigned* __restrict__ wsU, int numTiles) {
  __shared__ int tgt[2][TGT_INTS];                // 30720 B double buffer
  __shared__ float    redF[BLOCK / 32][5];
  __shared__ unsigned redU[BLOCK / 32][2];

  const int tid = threadIdx.x;
  // Low 32 bits of the flat address of an LDS object == its LDS offset
  // (LDS aperture is 4GB-aligned; HW adds the wave's LDS base to the vdst offset for async DMA).
  const uint32_t ldsBase = (uint32_t)(uintptr_t)(&tgt[0][0]);

  float pd[15];
#pragma unroll
  for (int j = 0; j < 15; ++j) pd[j] = pos_dist[j];   // uniform -> s_loads

  float s0 = 0.f, s1 = 0.f, s2 = 0.f, s3 = 0.f, s4 = 0.f;
  unsigned n1 = 0u, succ = 0u;

  int k = (int)blockIdx.x;
  if (k < numTiles) issueTile(targets, k, ldsBase, tid);   // prologue DMA

  int it = 0;
  for (; k < numTiles; k += (int)gridDim.x, ++it) {
    const int buf = it & 1;
    const int knext = k + (int)gridDim.x;

    asyncWait0();          // my wave's DMA for tile k has landed in LDS
    __syncthreads();       // all waves' DMA visible; prev reads of buf^1 done

    if (knext < numTiles)  // overlap next tile's DMA with this tile's compute
      issueTile(targets, knext,
                ldsBase + (uint32_t)((buf ^ 1) * TGT_INTS * 4), tid);

    const int r = k * TILE_ROWS + tid;

    // Read-once streams: NT hint keeps them from thrashing the 192MB L2.
    const float x0 = __builtin_nontemporal_load(out0 + r);
    const v4f o1a  = __builtin_nontemporal_load((const v4f*)out1 + (size_t)r * 2);
    const v4f o1b  = __builtin_nontemporal_load((const v4f*)out1 + (size_t)r * 2 + 1);
    const v4f o2   = __builtin_nontemporal_load((const v4f*)out2 + r);
    const float x3 = __builtin_nontemporal_load(out3 + r);
    const float x4 = __builtin_nontemporal_load(out4 + r);

    // stride-15 words, gcd(15,64)=1 -> bank-conflict-free across 32 lanes
    int t[15];
#pragma unroll
    for (int j = 0; j < 15; ++j) t[j] = tgt[buf][tid * 15 + j];

    // ---- head0 / head3 / head4 (binary, pos_weight = 1 - pos_dist[i]) ----
    binHead(x0, t[0],  1.f - pd[0],  s0, succ);
    binHead(x3, t[13], 1.f - pd[13], s3, succ);
    binHead(x4, t[14], 1.f - pd[14], s4, succ);

    // ---- head1: softmax probabilities, masked focal log + argmax match ----
    {
      float o[8] = {o1a.x, o1a.y, o1a.z, o1a.w, o1b.x, o1b.y, o1b.z, o1b.w};
      int aG = 0, bg = t[1];
      int aO = 0; float bo = o[0];
#pragma unroll
      for (int j = 1; j < 8; ++j) {               // first-occurrence argmax
        if (t[1 + j] > bg) { bg = t[1 + j]; aG = j; }
        if (o[j] > bo)     { bo = o[j];     aO = j; }
      }
      succ += (aG == aO) ? 1u : 0u;
#pragma unroll
      for (int j = 0; j < 8; ++j) {
        if (t[1 + j] == 1) {
          float om = 1.f - o[j];
          s1 += logf(o[j]) * om * om * (1.f - pd[1 + j]);
          n1 += 1u;
        }
      }
    }

    // ---- head2: weighted stable BCE on logits, succ on (logit > 0.5) ----
    {
      float xs[4] = {o2.x, o2.y, o2.z, o2.w};
#pragma unroll
      for (int j = 0; j < 4; ++j) {
        int g = t[9 + j];
        float x = xs[j];
        float bce = fmaxf(x, 0.f) - x * (float)g + log1pf(expf(-fabsf(x)));
        float w = g ? (1.f - pd[9 + j]) : pd[9 + j];
        s2 += w * bce;
        succ += ((((x > 0.5f) ? 1 : 0) == g) ? 1u : 0u);
      }
    }
  }

  // ---- block reduction: wave32 shfl tree, then LDS across 8 waves ----
  s0 = waveRedF(s0); s1 = waveRedF(s1); s2 = waveRedF(s2);
  s3 = waveRedF(s3); s4 = waveRedF(s4);
  n1 = waveRedU(n1); succ = waveRedU(succ);
  const int wave = tid >> 5, lane = tid & 31;
  if (lane == 0) {
    redF[wave][0] = s0; redF[wave][1] = s1; redF[wave][2] = s2;
    redF[wave][3] = s3; redF[wave][4] = s4;
    redU[wave][0] = n1; redU[wave][1] = succ;
  }
  __syncthreads();
  if (tid == 0) {
    float a0 = 0.f, a1 = 0.f, a2 = 0.f, a3 = 0.f, a4 = 0.f;
    unsigned u0 = 0u, u1 = 0u;
#pragma unroll
    for (int w = 0; w < BLOCK / 32; ++w) {
      a0 += redF[w][0]; a1 += redF[w][1]; a2 += redF[w][2];
      a3 += redF[w][3]; a4 += redF[w][4];
      u0 += redU[w][0]; u1 += redU[w][1];
    }
    const int g = (int)gridDim.x, b = (int)blockIdx.x;
    wsF[0 * g + b] = a0; wsF[1 * g + b] = a1; wsF[2 * g + b] = a2;
    wsF[3 * g + b] = a3; wsF[4 * g + b] = a4;
    wsU[0 * g + b] = u0; wsU[1 * g + b] = u1;
  }
}

__global__ __launch_bounds__(BLOCK) void pa_finalize(
    const float* __restrict__ wsF, const unsigned* __restrict__ wsU,
    float* __restrict__ out, int grid, float invB) {
  __shared__ float    rF[BLOCK / 32][5];
  __shared__ unsigned rU[BLOCK / 32][2];
  const int tid = threadIdx.x;
  float a[5] = {0.f, 0.f, 0.f, 0.f, 0.f};
  unsigned u[2] = {0u, 0u};
  for (int i = tid; i < grid; i += BLOCK) {
#pragma unroll
    for (int c = 0; c < 5; ++c) a[c] += wsF[c * grid + i];
    u[0] += wsU[0 * grid + i];
    u[1] += wsU[1 * grid + i];
  }
#pragma unroll
  for (int c = 0; c < 5; ++c) a[c] = waveRedF(a[c]);
  u[0] = waveRedU(u[0]); u[1] = waveRedU(u[1]);
  const int wave = tid >> 5, lane = tid & 31;
  if (lane == 0) {
#pragma unroll
    for (int c = 0; c < 5; ++c) rF[wave][c] = a[c];
    rU[wave][0] = u[0]; rU[wave][1] = u[1];
  }
  __syncthreads();
  if (tid == 0) {
    float s[5] = {0.f, 0.f, 0.f, 0.f, 0.f};
    unsigned n1 = 0u, succ = 0u;
#pragma unroll
    for (int w = 0; w < BLOCK / 32; ++w) {
#pragma unroll
      for (int c = 0; c < 5; ++c) s[c] += rF[w][c];
      n1 += rU[w][0]; succ += rU[w][1];
    }
    // binary heads: n = pos+neg = B always (targets in {0,1})
    float l0 = -s[0] * invB;
    float l1 = (n1 > 0u) ? (-s[1] / fmaxf((float)n1, 1.f)) : 0.f;
    float l2 = s[2] * invB * 0.25f;               // mean over [B,4]
    float l3 = -s[3] * invB;
    float l4 = -s[4] * invB;
    out[0] = l0 + l1 + l2 + l3 + l4;
    out[1] = (float)succ * 0.125f;                // / label_number (8)
  }
}

extern "C" void kernel_launch(void* const* d_in, const int* in_sizes, int n_in,
                              void* d_out, int out_size, void* d_ws,
                              size_t ws_size, hipStream_t stream) {
  const float* out0     = (const float*)d_in[0];
  const float* out1     = (const float*)d_in[1];
  const float* out2     = (const float*)d_in[2];
  const float* out3     = (const float*)d_in[3];
  const float* out4     = (const float*)d_in[4];
  const float* pos_dist = (const float*)d_in[5];
  const int*   targets  = (const int*)d_in[6];

  const int B        = in_sizes[0];          // rows (2^21, multiple of 256)
  const int numTiles = B / TILE_ROWS;
  int grid = numTiles < MAX_GRID ? numTiles : MAX_GRID;

  float*    wsF = (float*)d_ws;                       // 5*grid floats
  unsigned* wsU = (unsigned*)(wsF + (size_t)5 * grid); // 2*grid uints

  pa_loss_main<<<grid, BLOCK, 0, stream>>>(out0, out1, out2, out3, out4,
                                           pos_dist, targets, wsF, wsU,
                                           numTiles);
  pa_finalize<<<1, BLOCK, 0, stream>>>(wsF, wsU, (float*)d_out, grid,
                                       1.f / (float)B);
}